// CausalGraphLearner_67997922230391
// MI455X (gfx1250) — compile-verified
//
#include <hip/hip_runtime.h>

// Problem dims (fixed by reference)
#define B_ 16
#define V_ 128
#define H_ 128

typedef float v2f __attribute__((ext_vector_type(2)));
typedef float v8f __attribute__((ext_vector_type(8)));
typedef unsigned int u32x4 __attribute__((ext_vector_type(4)));
typedef int i32x8 __attribute__((ext_vector_type(8)));
typedef int i32x4 __attribute__((ext_vector_type(4)));

#if defined(__AMDGCN__) && __has_builtin(__builtin_amdgcn_tensor_load_to_lds)
#define USE_TDM 1
#else
#define USE_TDM 0
#endif

__device__ __forceinline__ float sigmoidf_(float x) {
    return 1.0f / (1.0f + __expf(-x));
}

#if USE_TDM
// Issue one TDM 2D tile load: 16 rows x 128 f32 (stride 128 f32) -> LDS.
// D# layout per CDNA5 ISA ch.8 (group0: flags/lds/global/type, group1: dims).
// This toolchain exposes the 6-arg builtin (g0, g1, g2, g3, g4, cpol).
__device__ __forceinline__ void tdm_load_tile16x128(unsigned lds_off,
                                                    const float* gptr) {
    unsigned long long ga = (unsigned long long)(uintptr_t)gptr;
    u32x4 g0;
    g0.x = 1u;                                      // count=1 valid descriptor
    g0.y = lds_off;                                 // LDS byte address
    g0.z = (unsigned)(ga & 0xffffffffu);            // global_addr[31:0]
    g0.w = (unsigned)((ga >> 32) & 0x01ffffffu)     // global_addr[56:32]
         | (2u << 30);                              // type = 2 ("image")
    i32x8 g1;
    g1[0] = 0x00020000;                             // data_size = 4B
    g1[1] = (int)(128u << 16);                      // tensor_dim0 = 128
    g1[2] = (int)(128u << 16);                      // tensor_dim1 = 128
    g1[3] = (int)(128u << 16);                      // tile_dim0 = 128
    g1[4] = 16;                                     // tile_dim1 = 16
    g1[5] = 128;                                    // tensor_dim0_stride = 128
    g1[6] = 0;
    g1[7] = 0;
    i32x4 z4 = {0, 0, 0, 0};
    i32x8 z8 = {0, 0, 0, 0, 0, 0, 0, 0};
    __builtin_amdgcn_tensor_load_to_lds(g0, g1, z4, z4, z8, 0);
}
#endif

// ---------------------------------------------------------------------------
// Kernel 1: adj[j,i] = sigmoid(L[j,i]) * (j!=i) * sigmoid((t[j]-t[i])*5)
// ---------------------------------------------------------------------------
__global__ void k_adj(const float* __restrict__ logits,
                      const float* __restrict__ torder,
                      float* __restrict__ adj_ws,
                      float* __restrict__ adj_out) {
    int idx = blockIdx.x * blockDim.x + threadIdx.x;   // 0 .. V*V-1
    int j = idx >> 7;
    int i = idx & (V_ - 1);
    float p = sigmoidf_(logits[idx]);
    p = (i == j) ? 0.0f : p;
    p *= sigmoidf_((torder[j] - torder[i]) * 5.0f);
    adj_ws[idx] = p;
    adj_out[idx] = p;
}

// colsum[i] = sum_j adj[j,i]   (for the b2 * adj.sum(0) bias term)
__global__ void k_colsum(const float* __restrict__ adj_ws,
                         float* __restrict__ colsum) {
    int i = threadIdx.x;
    float s = 0.0f;
    for (int j = 0; j < V_; ++j) s += adj_ws[j * V_ + i];
    colsum[i] = s;
}

// ---------------------------------------------------------------------------
// Kernel 2: cpb = X @ W1[:H] + b1 ; ep = X @ W1[H:]   (f32 WMMA 16x16x4)
// One 16-row tile per block; 8 waves, each owning one 16-wide h tile.
// ---------------------------------------------------------------------------
__global__ void __launch_bounds__(256)
k_in_gemm(const float* __restrict__ X,    // [B*V, H]
          const float* __restrict__ W1,   // [2H, H]
          const float* __restrict__ b1,   // [H]
          float* __restrict__ cpb,        // [B*V, H]
          float* __restrict__ ep)         // [B*V, H]
{
    const int r0   = blockIdx.x * 16;
    const int lane = threadIdx.x;          // 0..31
    const int h0   = threadIdx.y * 16;     // wave -> h tile
    const int m    = lane & 15;            // A row / B,C col
    const int koff = (lane >> 4) * 2;      // lanes 16..31 hold K+2

    v8f cc = {};   // cause accumulator
    v8f ce = {};   // effect accumulator

#pragma unroll 4
    for (int k0 = 0; k0 < H_; k0 += 4) {
        const int k = k0 + koff;
        v2f a = *(const v2f*)(X + (r0 + m) * H_ + k);   // 8B aligned (k even)
        v2f bc, be;
        bc.x = W1[(k    ) * H_ + h0 + m];
        bc.y = W1[(k + 1) * H_ + h0 + m];
        be.x = W1[(H_ + k    ) * H_ + h0 + m];
        be.y = W1[(H_ + k + 1) * H_ + h0 + m];
        cc = __builtin_amdgcn_wmma_f32_16x16x4_f32(false, a, false, bc,
                                                   (short)0, cc, false, false);
        ce = __builtin_amdgcn_wmma_f32_16x16x4_f32(false, a, false, be,
                                                   (short)0, ce, false, false);
    }

    const int mrow = (lane >> 4) * 8;      // C layout: lanes 16..31 -> M+8
#pragma unroll
    for (int r = 0; r < 8; ++r) {
        const int row = r0 + mrow + r;
        const int col = h0 + m;
        cpb[row * H_ + col] = cc[r] + b1[col];  // fold b1 into cause part
        ep [row * H_ + col] = ce[r];
    }
}

// ---------------------------------------------------------------------------
// Kernel 3 (fused): per (b, 16-row i-tile):
//   weighted[m,h] = sum_j adj[j,i0+m] * relu(ep[b,i0+m,h] + cpb[b,j,h])
//   effects tile  = weighted @ W2 + b2 * colsum          (f32 WMMA)
// cpb is streamed through LDS in 16-row chunks by the Tensor Data Mover,
// double-buffered (issue chunk c+1, s_wait_tensorcnt(1) => chunk c landed).
// ---------------------------------------------------------------------------
__global__ void __launch_bounds__(256)
k_fused(const float* __restrict__ cpb,     // [B*V, H]
        const float* __restrict__ ep,      // [B*V, H]
        const float* __restrict__ adj,     // [V, V]  adj[j,i]
        const float* __restrict__ colsum,  // [V]
        const float* __restrict__ W2,      // [H, H]
        const float* __restrict__ b2,      // [H]
        float* __restrict__ out)           // effects [B, V, H]
{
    __shared__ float s_ep[16 * 132];       // ep tile   [m][h]   (pad 132)
    __shared__ float s_at[16 * 132];       // adj^T     [m][j]
    __shared__ float s_w [16 * 132];       // weighted  [m][h]
    __shared__ float s_cp[2][16 * 128];    // TDM-staged cpb chunks [jj][h]

    const int b  = blockIdx.x >> 3;            // V/16 = 8 i-tiles
    const int i0 = (blockIdx.x & 7) * 16;
    const int t  = threadIdx.x;                // 0..255

    for (int e = t; e < 16 * H_; e += 256) {
        const int mm = e >> 7, hh = e & 127;
        s_ep[mm * 132 + hh] = ep[(b * V_ + i0 + mm) * H_ + hh];
    }
    for (int e = t; e < 16 * V_; e += 256) {
        const int mm = e >> 7, jj = e & 127;
        s_at[mm * 132 + jj] = adj[jj * V_ + i0 + mm];
    }
    // Warm caches for the final GEMM's B operand while the reduction runs.
    __builtin_prefetch(W2 + (t & 127) * H_, 0, 3);
    __builtin_prefetch(W2 + (t & 127) * H_ + 64, 0, 3);

    const float* cpb_b = cpb + (size_t)b * V_ * H_;
#if USE_TDM
    if (t == 0)
        tdm_load_tile16x128((unsigned)(uintptr_t)&s_cp[0][0], cpb_b);
#endif
    __syncthreads();

    // ---- fused relu + weighted j-reduction (8 i-rows per thread) ----
    const int h     = t & 127;
    const int mbase = (t >> 7) * 8;
    float epr[8], w[8];
#pragma unroll
    for (int mi = 0; mi < 8; ++mi) {
        epr[mi] = s_ep[(mbase + mi) * 132 + h];
        w[mi]   = 0.0f;
    }

    for (int c = 0; c < 8; ++c) {              // 8 chunks of 16 j-rows
        const int buf = c & 1;
#if USE_TDM
        if (t == 0) {
            if (c + 1 < 8) {
                tdm_load_tile16x128((unsigned)(uintptr_t)&s_cp[buf ^ 1][0],
                                    cpb_b + (c + 1) * 16 * H_);
                __builtin_amdgcn_s_wait_tensorcnt(1);  // chunk c complete
            } else {
                __builtin_amdgcn_s_wait_tensorcnt(0);
            }
        }
#else
        for (int e = t; e < 16 * H_; e += 256)
            s_cp[buf][e] = cpb_b[c * 16 * H_ + e];
#endif
        __syncthreads();                       // publish chunk c to all waves
#pragma unroll
        for (int jj = 0; jj < 16; ++jj) {
            const float cv = s_cp[buf][jj * 128 + h];
            const int j = c * 16 + jj;
#pragma unroll
            for (int mi = 0; mi < 8; ++mi) {
                const float hv = fmaxf(epr[mi] + cv, 0.0f);   // relu(ep+cp+b1)
                w[mi] = fmaf(s_at[(mbase + mi) * 132 + j], hv, w[mi]);
            }
        }
        __syncthreads();                       // buffer reuse fence
    }

#pragma unroll
    for (int mi = 0; mi < 8; ++mi)
        s_w[(mbase + mi) * 132 + h] = w[mi];
    __syncthreads();

    // ---- effects tile = weighted(16x128) @ W2(128x128), one 16x16 / wave --
    const int lane = t & 31;
    const int wave = t >> 5;
    const int n0   = wave * 16;
    const int m    = lane & 15;
    const int koff = (lane >> 4) * 2;

    v8f acc = {};
#pragma unroll 4
    for (int k0 = 0; k0 < H_; k0 += 4) {
        const int k = k0 + koff;
        v2f a, bb;
        a.x  = s_w[m * 132 + k];
        a.y  = s_w[m * 132 + k + 1];
        bb.x = W2[(k    ) * H_ + n0 + m];
        bb.y = W2[(k + 1) * H_ + n0 + m];
        acc = __builtin_amdgcn_wmma_f32_16x16x4_f32(false, a, false, bb,
                                                    (short)0, acc, false, false);
    }

    const int mrow = (lane >> 4) * 8;
#pragma unroll
    for (int r = 0; r < 8; ++r) {
        const int row = i0 + mrow + r;
        const int col = n0 + m;
        out[(b * V_ + row) * H_ + col] = acc[r] + b2[col] * colsum[row];
    }
}

// ---------------------------------------------------------------------------
extern "C" void kernel_launch(void* const* d_in, const int* in_sizes, int n_in,
                              void* d_out, int out_size, void* d_ws, size_t ws_size,
                              hipStream_t stream) {
    const float* variables = (const float*)d_in[0];   // [B,V,H]
    const float* logits    = (const float*)d_in[1];   // [V,V]
    const float* torder    = (const float*)d_in[2];   // [V]
    const float* W1        = (const float*)d_in[3];   // [2H,H]
    const float* b1        = (const float*)d_in[4];   // [H]
    const float* W2        = (const float*)d_in[5];   // [H,H]
    const float* b2        = (const float*)d_in[6];   // [H]

    float* out_eff = (float*)d_out;                   // [B,V,H]
    float* out_adj = out_eff + (size_t)B_ * V_ * H_;  // [V,V]

    float* ws     = (float*)d_ws;
    float* cpbuf  = ws;                               // [B*V, H]
    float* epbuf  = ws + (size_t)B_ * V_ * H_;        // [B*V, H]
    float* adjw   = ws + (size_t)2 * B_ * V_ * H_;    // [V, V]
    float* colsum = adjw + (size_t)V_ * V_;           // [V]

    k_adj<<<(V_ * V_) / 256, 256, 0, stream>>>(logits, torder, adjw, out_adj);
    k_colsum<<<1, V_, 0, stream>>>(adjw, colsum);

    dim3 blk2(32, 8);
    k_in_gemm<<<(B_ * V_) / 16, blk2, 0, stream>>>(variables, W1, b1, cpbuf, epbuf);

    k_fused<<<B_ * (V_ / 16), 256, 0, stream>>>(cpbuf, epbuf, adjw, colsum,
                                                W2, b2, out_eff);
}